// DTWLAYER_38448547233963
// MI455X (gfx1250) — compile-verified
//
#include <hip/hip_runtime.h>
#include <stdint.h>

typedef __attribute__((ext_vector_type(2))) float v2f;
typedef __attribute__((ext_vector_type(8))) float v8f;

#define BATCH 256
#define NF    64
#define NLEN  2048
#define MLEN  16
#define CHUNK 16
#define FPITCH 17                 // odd pitch -> conflict-free LDS banking
#define RPITCH (NF * FPITCH)      // 1088 floats per tile-row across all filters
#define BIG   1e30f

__global__ __launch_bounds__(64)
void DTW_cdna5_kernel(const float* __restrict__ x,
                      const float* __restrict__ kern,
                      float* __restrict__ out)
{
    __shared__ __align__(16) float xs[NLEN];           // staged x row (8 KB)
    __shared__ __align__(16) float ks[NF * MLEN];      // all kernels (4 KB)
    __shared__ float dbuf[CHUNK * RPITCH];             // WMMA d-tiles (68 KB)
    __shared__ float obuf[NF * FPITCH];                // output transpose (4.25 KB)

    const int tid  = threadIdx.x;         // 0..63, = filter id for DP phase
    const int b    = blockIdx.x;          // batch
    const int lane = tid & 31;
    const int wv   = tid >> 5;            // wave id 0/1
    const int cl   = lane & 15;           // column within 16-wide tile
    const bool hi  = lane >= 16;

    // ---- stage x[b,:] into LDS via CDNA5 async-DMA (ASYNCcnt path) ----
    {
        const float4* gx = (const float4*)(x + (size_t)b * NLEN);
        #pragma unroll
        for (int it = 0; it < (NLEN / 4) / 64; ++it) {   // 8 issues/thread
            int idx = it * 64 + tid;                      // float4 index
            unsigned ldsoff = (unsigned)(unsigned long long)(uintptr_t)(&xs[idx * 4]);
            const float4* g = gx + idx;
            asm volatile("global_load_async_to_lds_b128 %0, %1, off"
                         :: "v"(ldsoff), "v"(g) : "memory");
        }
        // kernels are tiny: plain vector loads
        const float4* gk  = (const float4*)kern;
        float4*       ks4 = (float4*)ks;
        #pragma unroll
        for (int it = 0; it < (NF * MLEN / 4) / 64; ++it)   // 4 issues/thread
            ks4[it * 64 + tid] = gk[it * 64 + tid];
        asm volatile("s_wait_asynccnt 0" ::: "memory");
    }
    __syncthreads();

    // ---- hoist per-filter WMMA B-operand pieces (chunk-invariant) ----
    // B (4x16): row0 = -2*k_j, row1 = 1, row2 = k_j^2, row3 = 0
    // layout mirrors A's K split: VGPR0 = rows {0,2}, VGPR1 = rows {1,3}
    float bxa[32];
    #pragma unroll
    for (int t = 0; t < 32; ++t) {
        float kv = ks[(wv * 32 + t) * MLEN + cl];
        bxa[t] = hi ? kv * kv : -2.0f * kv;
    }
    const float byc = hi ? 0.0f : 1.0f;

    float prev[MLEN];                 // D[i-1, 0..15] carried across chunks
    const int f = tid;

    for (int i0 = 0; i0 < NLEN; i0 += CHUNK) {
        // ===== producer phase: 32 WMMA d-tiles per wave =====
        // A (16x4): col0 = x_i, col1 = x_i^2, col2 = 1, col3 = 0
        float xv = xs[i0 + cl];
        v2f A;
        A.x = hi ? 1.0f : xv;         // lanes>=16 carry K=2 (=1) in VGPR0
        A.y = hi ? 0.0f : xv * xv;    // lanes>=16 carry K=3 (=0) in VGPR1
        const int rb = hi ? 8 : 0;
        #pragma unroll                 // FULL unroll: bxa[t] becomes a direct
        for (int t = 0; t < 32; ++t) { // register ref -> no m0 / v_movrels
            int ff = wv * 32 + t;
            v2f Bv;
            Bv.x = bxa[t];
            Bv.y = byc;
            v8f acc = {};
            // D[i,j] = x_i^2 - 2 x_i k_j + k_j^2  == (x_i - k_j)^2
            acc = __builtin_amdgcn_wmma_f32_16x16x4_f32(
                      false, A, false, Bv, (short)0, acc, false, false);
            #pragma unroll
            for (int v = 0; v < 8; ++v)
                dbuf[(rb + v) * RPITCH + ff * FPITCH + cl] = acc[v];
        }
        __syncthreads();   // d-tiles ready; prev obuf flush also complete

        // ===== DP phase: thread f advances its DTW by 16 rows =====
        #pragma unroll
        for (int r = 0; r < CHUNK; ++r) {
            const float* drow = &dbuf[r * RPITCH + f * FPITCH];
            float c;
            if (i0 == 0 && r == 0) {
                c = 0.0f;                         // row 0: cumulative sum
                #pragma unroll
                for (int j = 0; j < MLEN; ++j) { c += drow[j]; prev[j] = c; }
            } else {
                // j = 0: D[i,-1] and D[i-1,-1] are +inf -> min3 collapses
                float pm = prev[0];               // becomes D[i-1, j-1] for j=1
                c = drow[0] + pm;                 // D[i, 0] = d_0 + D[i-1, 0]
                prev[0] = c;
                #pragma unroll
                for (int j = 1; j < MLEN; ++j) {
                    float pj = prev[j];           // D[i-1, j]
                    float m  = fminf(fminf(pj, pm), c);   // -> v_min3_num_f32
                    c = drow[j] + m;              // D[i, j]
                    prev[j] = c;
                    pm = pj;
                }
            }
            obuf[f * FPITCH + r] = c;             // D[i, M-1]
        }
        __syncthreads();   // obuf ready; dbuf free for next producer phase

        // ===== coalesced output flush: 64B contiguous per 16 lanes =====
        float* ob = out + ((size_t)b * NF) * NLEN + i0;
        #pragma unroll
        for (int rep = 0; rep < CHUNK; ++rep) {
            int idx = rep * 64 + tid;
            int fo  = idx >> 4;
            int io  = idx & 15;
            ob[(size_t)fo * NLEN + io] = obuf[fo * FPITCH + io];
        }
        // no barrier needed: next chunk's first __syncthreads() orders
        // this flush's obuf reads before the next DP-phase obuf writes
    }
}

extern "C" void kernel_launch(void* const* d_in, const int* in_sizes, int n_in,
                              void* d_out, int out_size, void* d_ws, size_t ws_size,
                              hipStream_t stream) {
    const float* x    = (const float*)d_in[0];   // [256, 2048] f32
    const float* kern = (const float*)d_in[1];   // [64, 16]   f32
    float*       out  = (float*)d_out;           // [256, 64, 2048] f32
    (void)in_sizes; (void)n_in; (void)out_size; (void)d_ws; (void)ws_size;
    DTW_cdna5_kernel<<<BATCH, 64, 0, stream>>>(x, kern, out);
}